// Stylev2ALAEGeneratorBlock_67731634258361
// MI455X (gfx1250) — compile-verified
//
#include <hip/hip_runtime.h>
#include <hip/hip_bf16.h>
#include <math.h>

typedef __attribute__((ext_vector_type(2))) float v2f;
typedef __attribute__((ext_vector_type(8))) float v8f;

#define BN    4
#define CIN1  128
#define COUT  64
#define WSDIM 512
#define HI    128
#define WI    128
#define HO    256
#define WO    256
#define KPAD  12      // 9 taps zero-padded to 12 so K is a multiple of 4
#define CHB   8       // input channels staged per LDS block
#define PW    34      // staged patch width: 32 pixels + 1 halo each side
#define PELEM (3*PW)  // 102 floats per channel patch

constexpr float GAINC  = 1.3867504905630728f;          // sqrt(2/(1+0.2^2))
constexpr float HE1    = GAINC / 33.941125496954285f;  // GAIN/sqrt(128*9)
constexpr float HE2    = GAINC / 24.0f;                // GAIN/sqrt(64*9)
constexpr float HE_A   = 1.0f / 22.627416997969522f;   // 1/sqrt(512)
constexpr float HE_RGB = 0.03f / 8.0f;                 // 0.03/sqrt(64)
constexpr float EPSC   = 1e-8f;

// ---- workspace layout (float offsets) ----
#define OFF_S1  0
#define OFF_D1  (OFF_S1 + BN*CIN1)                 // 512
#define OFF_S2  (OFF_D1 + BN*COUT)                 // 768
#define OFF_D2  (OFF_S2 + BN*COUT)                 // 1024
#define OFF_W1F (OFF_D2 + BN*COUT)                 // 1280
#define OFF_W2F (OFF_W1F + BN*CIN1*COUT*KPAD)      // +393216
#define OFF_H1  (OFF_W2F + BN*COUT*COUT*KPAD)      // +196608
// total = 17,368,320 floats (~69.5 MB) of d_ws

// -------- styles + demod factors (tiny) --------
__global__ __launch_bounds__(128) void style_kernel(
    const float* __restrict__ w,   const float* __restrict__ a1w, const float* __restrict__ a1b,
    const float* __restrict__ a2w, const float* __restrict__ a2b,
    const float* __restrict__ c1w, const float* __restrict__ c2w,
    float* __restrict__ ws)
{
  __shared__ float wv[WSDIM];
  __shared__ float s1sh[CIN1];
  __shared__ float s2sh[COUT];
  const int b = blockIdx.x, tid = threadIdx.x;
  for (int k = tid; k < WSDIM; k += 128) wv[k] = w[b*WSDIM + k];
  __syncthreads();
  { // s1[b][i], i = tid (0..127)
    float acc = a1b[tid];
    for (int k = 0; k < WSDIM; ++k) acc += a1w[tid*WSDIM + k] * HE_A * wv[k];
    s1sh[tid] = acc;
    ws[OFF_S1 + b*CIN1 + tid] = acc;
  }
  if (tid < COUT) { // s2[b][o]
    float acc = a2b[tid];
    for (int k = 0; k < WSDIM; ++k) acc += a2w[tid*WSDIM + k] * HE_A * wv[k];
    s2sh[tid] = acc;
    ws[OFF_S2 + b*COUT + tid] = acc;
  }
  __syncthreads();
  if (tid < COUT) { // d1[b][o]
    float sum = 0.f;
    for (int i = 0; i < CIN1; ++i) {
      const float* p = c1w + (i*COUT + tid)*9;
      float wsq = 0.f;
      for (int t = 0; t < 9; ++t) { float v = p[t]*HE1; wsq += v*v; }
      sum += s1sh[i]*s1sh[i]*wsq;
    }
    ws[OFF_D1 + b*COUT + tid] = rsqrtf(sum + EPSC);
  } else { // d2[b][o], o = tid-64
    const int o = tid - 64;
    float sum = 0.f;
    for (int i = 0; i < COUT; ++i) {
      const float* p = c2w + (i*COUT + o)*9;
      float wsq = 0.f;
      for (int t = 0; t < 9; ++t) { float v = p[t]*HE2; wsq += v*v; }
      sum += s2sh[i]*s2sh[i]*wsq;
    }
    ws[OFF_D2 + b*COUT + o] = rsqrtf(sum + EPSC);
  }
}

// -------- fold modulated+demodulated (and flipped) weights: [b][i][o][KPAD] --------
__global__ void fold_kernel(const float* __restrict__ cw, float* __restrict__ ws,
                            int off_s, int off_d, int off_wf, int cin, float he)
{
  const int idx = blockIdx.x*blockDim.x + threadIdx.x;
  const int total = BN*cin*COUT*KPAD;
  if (idx >= total) return;
  const int t = idx % KPAD;
  const int o = (idx / KPAD) % COUT;
  const int i = (idx / (KPAD*COUT)) % cin;
  const int b =  idx / (KPAD*COUT*cin);
  float v = 0.f;
  if (t < 9) {
    const int r = t/3, s = t%3; // convt3 flips the kernel -> correlation with w[2-r][2-s]
    v = cw[((i*COUT + o)*3 + (2-r))*3 + (2-s)] * he
        * ws[off_s + b*cin + i] * ws[off_d + b*COUT + o] * GAINC;
  }
  ws[off_wf + idx] = v;
}

// -------- fused implicit-GEMM 3x3 conv via V_WMMA_F32_16X16X4_F32 --------
// grid = (WO/32, HO, B), block = 128 (4 waves). Wave m computes out-channels
// [16m,16m+16) x 32 pixels (two 16x16 D tiles sharing A-fragments) of one
// output row. Input patch (3 x 34, zero-padded; bilinear-upsampled when UP)
// is double-buffered in LDS: stage cb+1 while WMMAs consume cb.
template<int CINT, bool UP, bool DO_RGB>
__global__ __launch_bounds__(128) void conv_kernel(
    const float* __restrict__ src,  const float* __restrict__ wf,
    const float* __restrict__ noise,const float* __restrict__ nb,
    const float* __restrict__ bias, const float* __restrict__ rgbw,
    const float* __restrict__ rgbb, float* __restrict__ dst_h,
    float* __restrict__ dst_rgb)
{
  __shared__ float sh[2][CHB*PELEM];   // [buf][c][3 rows][34 cols]
  __shared__ float rgbpart[4][3][32];
  const int tid  = threadIdx.x;
  const int wave = tid >> 5, lane = tid & 31;
  const int n = lane & 15, half = lane >> 4;
  const int X0 = blockIdx.x * 32, Y = blockIdx.y, b = blockIdx.z;
  constexpr int NCB = CINT / CHB;

  // per-lane B-fragment LDS offsets for the 3 K-steps of each channel
  int boff[3][2];
  #pragma unroll
  for (int s3 = 0; s3 < 3; ++s3)
    #pragma unroll
    for (int v = 0; v < 2; ++v) {
      int k = s3*4 + 2*half + v;
      int t = k > 8 ? 8 : k;              // padded taps read a (finite) dummy; weight is 0
      boff[s3][v] = (t/3)*PW + (t%3) + n;
    }

  // stage one channel-block's 3x34 patches into LDS buffer `buf`
  auto stage = [&](int cb, int buf) {
    for (int idx = tid; idx < CHB*PELEM; idx += 128) {
      const int c = idx / PELEM, rem = idx % PELEM;
      const int r = rem / PW, col = rem % PW;
      const int jy = Y - 1 + r, jx = X0 - 1 + col;
      float v = 0.f;
      if (jy >= 0 && jy < HO && jx >= 0 && jx < WO) {
        const int ch = cb*CHB + c;
        if (UP) { // align-corners bilinear 128 -> 256, fused
          const float fy = jy * (127.0f/255.0f);
          const int   y0 = (int)fy;  const float wy = fy - (float)y0;
          const int   y1 = (y0+1 < HI) ? y0+1 : HI-1;
          const float fx = jx * (127.0f/255.0f);
          const int   x0 = (int)fx;  const float wx = fx - (float)x0;
          const int   x1 = (x0+1 < WI) ? x0+1 : WI-1;
          const float* p = src + (size_t)(b*CINT + ch)*HI*WI;
          const float t0 = p[y0*WI+x0]*(1.f-wx) + p[y0*WI+x1]*wx;
          const float t1 = p[y1*WI+x0]*(1.f-wx) + p[y1*WI+x1]*wx;
          v = t0*(1.f-wy) + t1*wy;
        } else {
          v = src[((size_t)(b*CINT + ch)*HO + jy)*WO + jx];
        }
      }
      sh[buf][idx] = v;
    }
  };

  v8f acc0 = {}, acc1 = {};
  const float* wbase = wf + ((size_t)(b*CINT)*COUT + (wave*16 + n))*KPAD + 2*half;

  stage(0, 0);
  __syncthreads();
  for (int cb = 0; cb < NCB; ++cb) {
    if (cb + 1 < NCB) stage(cb + 1, (cb + 1) & 1);   // prefetch next block
    const float* shb = sh[cb & 1];
    #pragma unroll
    for (int c = 0; c < CHB; ++c) {
      const float* wrow = wbase + (size_t)(cb*CHB + c)*COUT*KPAD;
      const float* shc  = shb + c*PELEM;
      #pragma unroll
      for (int s3 = 0; s3 < 3; ++s3) {
        v2f a = *(const v2f*)(wrow + s3*4);
        v2f bf0, bf1;
        bf0.x = shc[boff[s3][0]];      bf0.y = shc[boff[s3][1]];
        bf1.x = shc[boff[s3][0] + 16]; bf1.y = shc[boff[s3][1] + 16];
        acc0 = __builtin_amdgcn_wmma_f32_16x16x4_f32(
            false, a, false, bf0, (short)0, acc0, false, false);
        acc1 = __builtin_amdgcn_wmma_f32_16x16x4_f32(
            false, a, false, bf1, (short)0, acc1, false, false);
      }
    }
    __syncthreads();
  }

  // ---- epilogue: noise*baff + bias, leaky-ReLU(0.2), store; optional RGB 1x1 ----
  const float nz0 = noise[((size_t)b*HO + Y)*WO + X0 + n];
  const float nz1 = noise[((size_t)b*HO + Y)*WO + X0 + 16 + n];
  float rp0[3] = {0.f,0.f,0.f}, rp1[3] = {0.f,0.f,0.f};
  #pragma unroll
  for (int v = 0; v < 8; ++v) {
    const int oc = wave*16 + v + 8*half;      // C/D layout: VGPR v -> M=v(+8 for hi half)
    const float bo = bias[oc], nbo = nb[oc];
    float h0 = acc0[v] + nz0*nbo + bo;  h0 = fmaxf(h0, 0.2f*h0);
    float h1 = acc1[v] + nz1*nbo + bo;  h1 = fmaxf(h1, 0.2f*h1);
    float* dp = dst_h + ((size_t)(b*COUT + oc)*HO + Y)*WO + X0 + n;
    dp[0]  = h0;
    dp[16] = h1;
    if (DO_RGB) {
      #pragma unroll
      for (int j = 0; j < 3; ++j) {
        const float rw = rgbw[j*COUT + oc]*HE_RGB;
        rp0[j] += rw*h0;
        rp1[j] += rw*h1;
      }
    }
  }
  if (DO_RGB) {
    #pragma unroll
    for (int j = 0; j < 3; ++j) {
      rp0[j] += __shfl_xor(rp0[j], 16, 32);
      rp1[j] += __shfl_xor(rp1[j], 16, 32);
    }
    if (half == 0) {
      #pragma unroll
      for (int j = 0; j < 3; ++j) {
        rgbpart[wave][j][n]      = rp0[j];
        rgbpart[wave][j][16 + n] = rp1[j];
      }
    }
    __syncthreads();
    if (tid < 96) {
      const int j = tid / 32, p = tid % 32;
      const float s = rgbpart[0][j][p] + rgbpart[1][j][p]
                    + rgbpart[2][j][p] + rgbpart[3][j][p] + rgbb[j];
      dst_rgb[((size_t)(b*3 + j)*HO + Y)*WO + X0 + p] = s;
    }
  }
}

extern "C" void kernel_launch(void* const* d_in, const int* in_sizes, int n_in,
                              void* d_out, int out_size, void* d_ws, size_t ws_size,
                              hipStream_t stream) {
  (void)in_sizes; (void)n_in; (void)out_size; (void)ws_size;
  const float* x       = (const float*)d_in[0];
  const float* wsty    = (const float*)d_in[1];
  const float* conv1_w = (const float*)d_in[2];
  const float* conv2_w = (const float*)d_in[3];
  const float* bias1   = (const float*)d_in[4];
  const float* bias2   = (const float*)d_in[5];
  const float* a1w     = (const float*)d_in[6];
  const float* a1b     = (const float*)d_in[7];
  const float* a2w     = (const float*)d_in[8];
  const float* a2b     = (const float*)d_in[9];
  const float* baff1   = (const float*)d_in[10];
  const float* baff2   = (const float*)d_in[11];
  const float* rgbw    = (const float*)d_in[12];
  const float* rgbb    = (const float*)d_in[13];
  const float* noise1  = (const float*)d_in[14];
  const float* noise2  = (const float*)d_in[15];

  float* ws     = (float*)d_ws;
  float* out_h  = (float*)d_out;
  float* out_rgb= out_h + (size_t)BN*COUT*HO*WO;

  // 1) styles + demod factors
  style_kernel<<<BN, 128, 0, stream>>>(wsty, a1w, a1b, a2w, a2b, conv1_w, conv2_w, ws);

  // 2) fold per-batch modulated/demodulated/flipped weights (K padded to 12)
  {
    const int tot1 = BN*CIN1*COUT*KPAD;
    fold_kernel<<<(tot1+255)/256, 256, 0, stream>>>(conv1_w, ws, OFF_S1, OFF_D1, OFF_W1F, CIN1, HE1);
    const int tot2 = BN*COUT*COUT*KPAD;
    fold_kernel<<<(tot2+255)/256, 256, 0, stream>>>(conv2_w, ws, OFF_S2, OFF_D2, OFF_W2F, COUT, HE2);
  }

  // 3) conv1: fused bilinear-upsample + modulated 3x3 (128->64) + noise/bias/lrelu -> h1 (L2-resident)
  conv_kernel<CIN1, true, false><<<dim3(WO/32, HO, BN), 128, 0, stream>>>(
      x, ws + OFF_W1F, noise1, baff1, bias1, nullptr, nullptr, ws + OFF_H1, nullptr);

  // 4) conv2: modulated 3x3 (64->64) + noise/bias/lrelu -> h, fused RGB 1x1 head -> rgb
  conv_kernel<COUT, false, true><<<dim3(WO/32, HO, BN), 128, 0, stream>>>(
      ws + OFF_H1, ws + OFF_W2F, noise2, baff2, bias2, rgbw, rgbb, out_h, out_rgb);
}